// BiNCPModel_81346680586939
// MI455X (gfx1250) — compile-verified
//
#include <hip/hip_runtime.h>

typedef float v2f __attribute__((ext_vector_type(2)));
typedef float v8f __attribute__((ext_vector_type(8)));

#define NWAVES 8

// ---------------- geometry ----------------
// IN=64, LAYERS=[135,89,32]; layer l consumes z_l, contiguous in LDS:
//   z0 = [x(64)   | h0(136)]  FINP=200 pitch 201
//   z1 = [h0(136) | h1(92) ]  FINP=228 pitch 229
//   z2 = [h1(92)  | h2(32) ]  FINP=124 pitch 125
// NPAD (N padded to 16): 144 / 96 / 32 ; NT tiles: 9 / 6 / 2
constexpr int WSZ0 = 200 * 144;
constexpr int WSZ1 = 228 * 96;
constexpr int WSZ2 = 124 * 32;
constexpr int WTOT = 3 * (WSZ0 + WSZ1 + WSZ2);   // 163968 floats padded weights / dir
constexpr int BTOT = 3 * (144 + 96 + 32);        // 816 floats padded biases / dir
constexpr int PERDIR = WTOT + BTOT;              // 164784
constexpr int DIROUT_OFF = 2 * PERDIR;           // [2][256][512][32] buffer
constexpr int WL1 = 3 * WSZ0;
constexpr int WL2 = WL1 + 3 * WSZ1;
constexpr int BL1 = 3 * 144;
constexpr int BL2 = BL1 + 3 * 96;

// branch-free activations on the v_exp_f32 / v_rcp_f32 fast path
__device__ __forceinline__ float fast_tanh(float x) {
    return 1.0f - 2.0f * __builtin_amdgcn_rcpf(__expf(2.0f * x) + 1.0f);
}
__device__ __forceinline__ float fast_sig(float x) {
    return __builtin_amdgcn_rcpf(1.0f + __expf(-x));
}

// ---------------- weight/bias pre-pack ----------------
// Weights packed in WMMA B-fragment order [mat][ntile][kchunk][lane][2] so the
// inner loop is one coalesced global_load_b64 per matrix per k-step.
// Folds: w1*mask, w2*mask, (wa+wb); zero pad; gate bias = ba+bb.
__global__ __launch_bounds__(256) void prep_layer(
    const float* __restrict__ w1, const float* __restrict__ w2,
    const float* __restrict__ wa, const float* __restrict__ wb,
    const float* __restrict__ b1, const float* __restrict__ b2,
    const float* __restrict__ ba, const float* __restrict__ bb,
    const float* __restrict__ mask,
    float* __restrict__ wdst, float* __restrict__ bdst,
    int prev, int h, int finp, int p0, int npad)
{
    const int kc_cnt = finp >> 2;
    const int perTile = kc_cnt * 64;
    const int wtot = finp * npad;
    const int total = 3 * wtot + 3 * npad;
    for (int i = blockIdx.x * blockDim.x + threadIdx.x; i < total;
         i += gridDim.x * blockDim.x) {
        if (i < 3 * wtot) {
            const int mat = i / wtot;
            const int e = i - mat * wtot;
            const int nt = e / perTile;
            const int r2 = e - nt * perTile;
            const int kc = r2 >> 6;
            const int q = r2 & 63;
            const int lane = q >> 1;
            const int j = q & 1;
            const int row = (kc << 2) + ((lane >> 4) << 1) + j;  // logical K
            const int col = nt * 16 + (lane & 15);               // logical N
            float v = 0.0f;
            if (col < h) {
                int sr = -1;
                if (row < prev) sr = row;                                   // input part
                else if (row >= p0 && row < p0 + h) sr = prev + (row - p0); // recurrent part
                if (sr >= 0) {
                    const int si = sr * h + col;
                    if (mat == 0)      v = w1[si] * mask[si];
                    else if (mat == 1) v = w2[si] * mask[si];
                    else               v = wa[si] + wb[si];                 // gate fold
                }
            }
            wdst[i] = v;
        } else {
            const int jj = i - 3 * wtot;
            const int mat = jj / npad;
            const int c = jj - mat * npad;
            float v = 0.0f;
            if (c < h)
                v = (mat == 0) ? b1[c] : (mat == 1) ? b2[c] : (ba[c] + bb[c]);
            bdst[mat * npad + c] = v;
        }
    }
}

// ---------------- one CfC layer step ----------------
// Contiguous z source => A-fragment reads are immediate-offset ds_load_2addr
// from one loop-invariant base. Biases preloaded in registers (b1v/b2v/btv[ROUNDS]).
template <int FINP, int NPAD, int NT, int HS, int P0, int PITCHZ, int PITCHN,
          bool HASNEXT, bool LAST, int ROUNDS>
__device__ __forceinline__ void layer_step(
    const float* __restrict__ wbase,
    const float* __restrict__ zsrc, float* __restrict__ zself, float* __restrict__ znext,
    const float* b1v, const float* b2v, const float* btv,
    int wave, int lane, float* __restrict__ goutRow, int tileRow)
{
    constexpr int WSZ = FINP * NPAD;
    constexpr int KC = FINP >> 2;
    const int m = lane & 15;
    const int koff = (lane >> 4) << 1;
    const float* abase = zsrc + m * PITCHZ + koff;

    v8f hn[ROUNDS];

#pragma unroll
    for (int rd = 0; rd < ROUNDS; ++rd) {
        const int nt = wave + rd * NWAVES;
        if (nt < NT) {                       // wave-uniform: EXEC stays all-ones
            v8f c0 = {0.f,0.f,0.f,0.f,0.f,0.f,0.f,0.f};
            v8f c1 = {0.f,0.f,0.f,0.f,0.f,0.f,0.f,0.f};
            v8f c2 = {0.f,0.f,0.f,0.f,0.f,0.f,0.f,0.f};
            const float* wp0 = wbase + (size_t)nt * (KC * 64) + (lane << 1);
            const float* wp1 = wp0 + WSZ;
            const float* wp2 = wp1 + WSZ;
            for (int kc = 0; kc < KC; ++kc) {
                const int kb = kc << 2;
                v2f a; a.x = abase[kb]; a.y = abase[kb + 1];     // ds_load_2addr, imm offsets
                const v2f b0 = *(const v2f*)(wp0 + kc * 64);     // coalesced b64 fragments
                const v2f b1 = *(const v2f*)(wp1 + kc * 64);
                const v2f b2 = *(const v2f*)(wp2 + kc * 64);
                c0 = __builtin_amdgcn_wmma_f32_16x16x4_f32(false, a, false, b0, (short)0, c0, false, false);
                c1 = __builtin_amdgcn_wmma_f32_16x16x4_f32(false, a, false, b1, (short)0, c1, false, false);
                c2 = __builtin_amdgcn_wmma_f32_16x16x4_f32(false, a, false, b2, (short)0, c2, false, false);
            }
            v8f h;
#pragma unroll
            for (int r = 0; r < 8; ++r) {
                const float f1 = fast_tanh(c0[r] + b1v[rd]);
                const float f2 = fast_tanh(c1[r] + b2v[rd]);
                const float tg = fast_sig(c2[r] + btv[rd]);
                h[r] = f1 + tg * (f2 - f1);   // ff1*(1-t) + t*ff2
            }
            hn[rd] = h;
        }
    }

    __syncthreads();  // all reads of old z complete

    const int rbase = (lane >> 4) << 3;
#pragma unroll
    for (int rd = 0; rd < ROUNDS; ++rd) {
        const int nt = wave + rd * NWAVES;
        if (nt < NT) {
            const int col = nt * 16 + m;
#pragma unroll
            for (int r = 0; r < 8; ++r) {
                const int row = rbase + r;
                if (col < HS) {
                    zself[row * PITCHZ + P0 + col] = hn[rd][r];        // own recurrent slot
                    if (HASNEXT) znext[row * PITCHN + col] = hn[rd][r]; // next layer's input slot
                }
                if (LAST) {   // motor layer -> dir-output buffer (col < 32 always)
                    goutRow[(size_t)(tileRow + row) * (512 * 32) + col] = hn[rd][r];
                }
            }
        }
    }
    __syncthreads();  // new z visible before next consumer
}

// ---------------- main recurrent kernel: 1 block = (direction, 16-row batch tile) ----------------
__global__ __launch_bounds__(256) void ncp_rnn_kernel(
    const float* __restrict__ x, const float* __restrict__ wsw, float* __restrict__ dirout)
{
    __shared__ float z0[16 * 201];   // [x | h0], odd pitch: conflict-free column reads
    __shared__ float z1[16 * 229];   // [h0 | h1]
    __shared__ float z2[16 * 125];   // [h1 | h2]

    const int dir = blockIdx.x >> 4;
    const int tileRow = (blockIdx.x & 15) << 4;
    const int tid = threadIdx.x;
    const int wave = tid >> 5;
    const int lane = tid & 31;
    const int m = lane & 15;

    for (int i = tid; i < 16 * 201; i += 256) z0[i] = 0.0f;
    for (int i = tid; i < 16 * 229; i += 256) z1[i] = 0.0f;
    for (int i = tid; i < 16 * 125; i += 256) z2[i] = 0.0f;

    const float* wd = wsw + (size_t)dir * PERDIR;
    const float* w0 = wd;               const float* bb0 = wd + WTOT;
    const float* w1 = wd + WL1;         const float* bb1 = wd + WTOT + BL1;
    const float* w2 = wd + WL2;         const float* bb2 = wd + WTOT + BL2;
    float* goutD = dirout + (size_t)dir * (256u * 512u * 32u);

    // ---- hoist biases (loop-invariant over t) into registers ----
    float b1L0[2], b2L0[2], btL0[2];
#pragma unroll
    for (int rd = 0; rd < 2; ++rd) {
        const int nt = wave + rd * NWAVES;
        const int nc = nt * 16 + m;
        const bool a = nt < 9;
        b1L0[rd] = a ? bb0[nc] : 0.0f;
        b2L0[rd] = a ? bb0[144 + nc] : 0.0f;
        btL0[rd] = a ? bb0[288 + nc] : 0.0f;
    }
    float b1L1[1], b2L1[1], btL1[1];
    {
        const int nc = wave * 16 + m;
        const bool a = wave < 6;
        b1L1[0] = a ? bb1[nc] : 0.0f;
        b2L1[0] = a ? bb1[96 + nc] : 0.0f;
        btL1[0] = a ? bb1[192 + nc] : 0.0f;
    }
    float b1L2[1], b2L2[1], btL2[1];
    {
        const int nc = wave * 16 + m;
        const bool a = wave < 2;
        b1L2[0] = a ? bb2[nc] : 0.0f;
        b2L2[0] = a ? bb2[32 + nc] : 0.0f;
        btL2[0] = a ? bb2[64 + nc] : 0.0f;
    }

    // ---- software-pipelined x tile: prefetch t+1 during step t ----
    const int xrow = tid >> 6;          // 4 rows per pass, 4 passes via +4 rows
    const int xcol = tid & 63;
    float px[4];
    {
        const int tsrc0 = dir ? 511 : 0;
#pragma unroll
        for (int j = 0; j < 4; ++j)
            px[j] = x[((size_t)(tileRow + xrow + 4 * j) * 512 + tsrc0) * 64 + xcol];
    }

    for (int t = 0; t < 512; ++t) {
        const int tsrc = dir ? (511 - t) : t;  // bwd consumes reversed x; output lands at tsrc
#pragma unroll
        for (int j = 0; j < 4; ++j)
            z0[(xrow + 4 * j) * 201 + xcol] = px[j];
        if (t < 511) {                          // issue next tile's loads; overlap with compute
            const int tn = dir ? (510 - t) : (t + 1);
#pragma unroll
            for (int j = 0; j < 4; ++j)
                px[j] = x[((size_t)(tileRow + xrow + 4 * j) * 512 + tn) * 64 + xcol];
        }
        __syncthreads();
        layer_step<200, 144, 9, 136,  64, 201, 229, true,  false, 2>(
            w0, z0, z0, z1, b1L0, b2L0, btL0, wave, lane, nullptr, tileRow);
        layer_step<228,  96, 6,  92, 136, 229, 125, true,  false, 1>(
            w1, z1, z1, z2, b1L1, b2L1, btL1, wave, lane, nullptr, tileRow);
        layer_step<124,  32, 2,  32,  92, 125,   1, false, true,  1>(
            w2, z2, z2, nullptr, b1L2, b2L2, btL2, wave, lane,
            goutD + (size_t)tsrc * 32, tileRow);
    }
}

// ---------------- final FC: out[b,t,:] = [f|b] @ fc_w + fc_b, via WMMA ----------------
__global__ __launch_bounds__(256) void fc_kernel(
    const float* __restrict__ dirout, const float* __restrict__ fcw,
    const float* __restrict__ fcb, float* __restrict__ out)
{
    const int lane = threadIdx.x & 31;
    const int wave = threadIdx.x >> 5;
    const size_t mt = (size_t)blockIdx.x * 8 + wave;   // 8192 row tiles over B*T=131072
    const int m = lane & 15;
    const int koff = (lane >> 4) << 1;
    const size_t rg = mt * 16 + m;
    const float* f0 = dirout + rg * 32;
    const float* f1 = dirout + (size_t)(256u * 512u * 32u) + rg * 32;
    const int rbase = (lane >> 4) << 3;
#pragma unroll
    for (int nt = 0; nt < 2; ++nt) {
        const int ncol = nt * 16 + m;
        v8f c = {0.f,0.f,0.f,0.f,0.f,0.f,0.f,0.f};
        for (int k = 0; k < 32; k += 4) {
            const int ka = k + koff;            // even -> 8B aligned b64 loads
            const v2f a  = *(const v2f*)(f0 + ka);
            v2f b;  b.x = fcw[ka * 32 + ncol];        b.y = fcw[(ka + 1) * 32 + ncol];
            c = __builtin_amdgcn_wmma_f32_16x16x4_f32(false, a, false, b, (short)0, c, false, false);
            const v2f a2 = *(const v2f*)(f1 + ka);
            v2f b2; b2.x = fcw[(32 + ka) * 32 + ncol]; b2.y = fcw[(33 + ka) * 32 + ncol];
            c = __builtin_amdgcn_wmma_f32_16x16x4_f32(false, a2, false, b2, (short)0, c, false, false);
        }
        const float bias = fcb[ncol];
#pragma unroll
        for (int r = 0; r < 8; ++r) {
            out[(mt * 16 + rbase + r) * 32 + ncol] = c[r] + bias;
        }
    }
}

// ---------------- host launcher ----------------
extern "C" void kernel_launch(void* const* d_in, const int* in_sizes, int n_in,
                              void* d_out, int out_size, void* d_ws, size_t ws_size,
                              hipStream_t stream) {
    (void)in_sizes; (void)n_in; (void)out_size; (void)ws_size;
    // d_in: 0:x, 1..24:params_fwd (3 layers x [w1,b1,w2,b2,wa,ba,wb,bb]),
    // 25..48:params_bwd, 49:fc_w, 50:fc_b, 51..53:masks
    const float* x    = (const float*)d_in[0];
    const float* fc_w = (const float*)d_in[49];
    const float* fc_b = (const float*)d_in[50];
    float* wsf = (float*)d_ws;
    float* dirout = wsf + DIROUT_OFF;

    const int PREVh[3] = {64, 135, 89};
    const int Hh[3]    = {135, 89, 32};
    const int FINPh[3] = {200, 228, 124};
    const int P0h[3]   = {64, 136, 92};
    const int NPADh[3] = {144, 96, 32};
    const int WLOFFh[3] = {0, WL1, WL2};
    const int BLOFFh[3] = {0, BL1, BL2};

    for (int dir = 0; dir < 2; ++dir) {
        for (int l = 0; l < 3; ++l) {
            const int base = 1 + dir * 24 + l * 8;
            prep_layer<<<128, 256, 0, stream>>>(
                (const float*)d_in[base + 0], (const float*)d_in[base + 2],
                (const float*)d_in[base + 4], (const float*)d_in[base + 6],
                (const float*)d_in[base + 1], (const float*)d_in[base + 3],
                (const float*)d_in[base + 5], (const float*)d_in[base + 7],
                (const float*)d_in[51 + l],
                wsf + (size_t)dir * PERDIR + WLOFFh[l],
                wsf + (size_t)dir * PERDIR + WTOT + BLOFFh[l],
                PREVh[l], Hh[l], FINPh[l], P0h[l], NPADh[l]);
        }
    }
    ncp_rnn_kernel<<<32, 256, 0, stream>>>(x, wsf, dirout);
    fc_kernel<<<1024, 256, 0, stream>>>(dirout, fc_w, fc_b, (float*)d_out);
}